// DistogramHead_91207925498226
// MI455X (gfx1250) — compile-verified
//
#include <hip/hip_runtime.h>
#include <math.h>

typedef __attribute__((ext_vector_type(2))) float v2f;
typedef __attribute__((ext_vector_type(4))) float f4;
typedef __attribute__((ext_vector_type(8))) float v8f;

#define DM   512   // d_model
#define NN   512   // sequence length
#define BINS 64

// exact GELU, matching torch nn.GELU default / jax.nn.gelu(approximate=False)
__device__ __forceinline__ float gelu_exact(float x) {
  return 0.5f * x * (1.0f + erff(x * 0.70710678118654752440f));
}

// ---------------------------------------------------------------------------
// Kernel 1: hi = x @ W1[:d] + b1 (bias folded),  hj = x @ W1[d:]
// One 16x16 output tile per wave, V_WMMA_F32_16X16X4_F32 over K=512.
// ---------------------------------------------------------------------------
__global__ __launch_bounds__(256) void hij_gemm_kernel(
    const float* __restrict__ x, const float* __restrict__ W1,
    const float* __restrict__ b1, float* __restrict__ hi,
    float* __restrict__ hj) {
  const float* W  = W1 + (size_t)blockIdx.z * DM * DM;   // z=0: W1[:d], z=1: W1[d:]
  float* out      = blockIdx.z ? hj : hi;

  const int lane = threadIdx.x & 31;
  const int w    = threadIdx.x >> 5;                      // wave in block (0..7)
  const int r0   = (blockIdx.y * 8 + w) * 16;             // row tile
  const int c0   = blockIdx.x * 16;                       // col tile
  const int m    = lane & 15;
  const int kh   = (lane >> 4) * 2;                       // K sub-offset per lane half

  v8f acc = {};
  for (int k0 = 0; k0 < DM; k0 += 4) {
    const int kk = k0 + kh;
    // A 16x4: lane m = row, VGPR0/1 = K = kk, kk+1
    v2f a = *(const v2f*)(x + (r0 + m) * DM + kk);
    // B 4x16: row-striped across lanes; VGPR0 = row kk, VGPR1 = row kk+1
    v2f b;
    b.x = W[(size_t)kk * DM + c0 + m];
    b.y = W[(size_t)(kk + 1) * DM + c0 + m];
    acc = __builtin_amdgcn_wmma_f32_16x16x4_f32(false, a, false, b,
                                                (short)0, acc, false, false);
  }
  // fold b1 into hi so the N^2*d hot loop never touches it
  const float bias = (blockIdx.z == 0) ? b1[c0 + m] : 0.0f;
  // C/D layout: VGPR r -> M = r + 8*(lane>=16), N = lane&15
  const int rbase = r0 + 8 * (lane >> 4);
#pragma unroll
  for (int r = 0; r < 8; ++r)
    out[(size_t)(rbase + r) * DM + c0 + m] = acc[r] + bias;
}

// ---------------------------------------------------------------------------
// Kernel 2: fused pair-add + exact GELU + (x W2 + b2) projection.
// Workgroup: 512 threads = 16 waves; block covers 16 i x 16 j pairs.
// Wave w owns i = i0+w and computes the 16(j) x 64(bins) tile with four
// V_WMMA_F32_16X16X4_F32 accumulators, K = 512.
//
// LDS:
//  - sW2p: W2 repacked as row pairs {W2[2k2][e], W2[2k2+1][e]} with a
//    160-dword pair-row stride. The stride rotates the bank origin by 32
//    per pair-row, so the two lane halves (which read adjacent pair-rows)
//    use disjoint bank halves -> conflict-free ds_load_b64 B fragments.
//  - sHj: 16 hj rows, staged with CDNA5 async global->LDS DMA (ASYNCcnt),
//    overlapped with the VGPR-staged W2 repack.
// ---------------------------------------------------------------------------
#define W2P_STRIDE 160   // dwords per pair-row (128 data + 32 rotation pad)

__global__ __launch_bounds__(512) void disto_main_kernel(
    const float* __restrict__ hi_g, const float* __restrict__ hj_g,
    const float* __restrict__ W2_g, const float* __restrict__ b2_g,
    float* __restrict__ out) {
  __shared__ float sW2p[(DM / 2) * W2P_STRIDE];   // 160 KB
  __shared__ float sHj[16 * DM];                  // 32 KB

  const int tid = threadIdx.x;       // 0..511
  const int j0  = blockIdx.x * 16;
  const int i0  = blockIdx.y * 16;

  // ---- issue async DMA of the 16 hj rows (32KB contiguous) into LDS ----
  {
    const char* src = (const char*)(hj_g + (size_t)j0 * DM);
    const unsigned ldsbase = (unsigned)(uintptr_t)(void*)sHj;  // addr[31:0] = LDS offset
#pragma unroll
    for (int q = 0; q < 4; ++q) {
      const int c = tid + q * 512;                 // 16B chunk id (2048 total)
      unsigned loff = ldsbase + c * 16;
      unsigned long long ga = (unsigned long long)(src + c * 16);
      asm volatile("global_load_async_to_lds_b128 %0, %1, off"
                   :: "v"(loff), "v"(ga) : "memory");
    }
  }

  // ---- repack W2 into pair-interleaved LDS layout (overlaps the DMA) ----
#pragma unroll
  for (int q = 0; q < 32; ++q) {
    const int idx = tid + q * 512;                 // 0..16383
    const int k2  = idx >> 6;                      // pair-row
    const int e   = idx & 63;                      // bin
    v2f v;
    v.x = W2_g[(size_t)(2 * k2) * BINS + e];
    v.y = W2_g[(size_t)(2 * k2 + 1) * BINS + e];
    *(v2f*)(sW2p + k2 * W2P_STRIDE + 2 * e) = v;
  }

  asm volatile("s_wait_asynccnt 0x0" ::: "memory");
  __syncthreads();

  const int lane = tid & 31;
  const int w    = tid >> 5;          // wave id -> local i
  const int i    = i0 + w;
  const int m    = lane & 15;         // A row (= local j), B/D column
  const int kh   = (lane >> 4) * 2;

  const float* hir = hi_g + (size_t)i * DM;   // one row (b1 pre-folded), hot in WGP$/L2
  const float* hjr = sHj + m * DM;

  v8f acc0 = {}, acc1 = {}, acc2 = {}, acc3 = {};
  for (int k0 = 0; k0 < DM; k0 += 4) {
    const int kk = k0 + kh;
    // fused A fragment: gelu(hi[i,k]+b1[k] + hj[j,k])
    v2f gv = *(const v2f*)(hir + kk);           // global b64 (cached)
    v2f hv = *(const v2f*)(hjr + kk);           // ds b64
    v2f a;
    a.x = gelu_exact(gv.x + hv.x);
    a.y = gelu_exact(gv.y + hv.y);
    // B fragments: conflict-free ds_load_b64 from pair-packed W2
    const float* wp = sW2p + (kk >> 1) * W2P_STRIDE;
    v2f bb0 = *(const v2f*)(wp + 2 * m);
    v2f bb1 = *(const v2f*)(wp + 2 * (16 + m));
    v2f bb2 = *(const v2f*)(wp + 2 * (32 + m));
    v2f bb3 = *(const v2f*)(wp + 2 * (48 + m));
    acc0 = __builtin_amdgcn_wmma_f32_16x16x4_f32(false, a, false, bb0, (short)0, acc0, false, false);
    acc1 = __builtin_amdgcn_wmma_f32_16x16x4_f32(false, a, false, bb1, (short)0, acc1, false, false);
    acc2 = __builtin_amdgcn_wmma_f32_16x16x4_f32(false, a, false, bb2, (short)0, acc2, false, false);
    acc3 = __builtin_amdgcn_wmma_f32_16x16x4_f32(false, a, false, bb3, (short)0, acc3, false, false);
  }

  // ---- epilogue: add b2, store. D layout: VGPR r -> M=r+8*(lane>=16), N=lane&15
  const float bias0 = b2_g[m];
  const float bias1 = b2_g[16 + m];
  const float bias2 = b2_g[32 + m];
  const float bias3 = b2_g[48 + m];
  const int jbase = j0 + 8 * (lane >> 4);
#pragma unroll
  for (int r = 0; r < 8; ++r) {
    float* o = out + ((size_t)i * NN + (jbase + r)) * BINS;
    o[m]      = acc0[r] + bias0;
    o[16 + m] = acc1[r] + bias1;
    o[32 + m] = acc2[r] + bias2;
    o[48 + m] = acc3[r] + bias3;
  }
}

// ---------------------------------------------------------------------------
extern "C" void kernel_launch(void* const* d_in, const int* in_sizes, int n_in,
                              void* d_out, int out_size, void* d_ws, size_t ws_size,
                              hipStream_t stream) {
  const float* x  = (const float*)d_in[0];   // [1,512,512]
  const float* W1 = (const float*)d_in[1];   // [1024,512]
  const float* b1 = (const float*)d_in[2];   // [512]
  const float* W2 = (const float*)d_in[3];   // [512,64]
  const float* b2 = (const float*)d_in[4];   // [64]
  float* out = (float*)d_out;                // [512,512,64]

  float* hi = (float*)d_ws;                  // 512*512 f32 (b1 folded in)
  float* hj = hi + (size_t)NN * DM;          // 512*512 f32  (2 MB total)

  dim3 g1(NN / 16, 4, 2);                    // 32 col tiles x 4 row blocks x {hi,hj}
  hij_gemm_kernel<<<g1, 256, 0, stream>>>(x, W1, b1, hi, hj);

  dim3 g2(NN / 16, NN / 16);                 // 32 x 32 pair blocks
  disto_main_kernel<<<g2, 512, 0, stream>>>(hi, hj, W2, b2, out);
}